// DiscreteBKI_26216480375243
// MI455X (gfx1250) — compile-verified
//
#include <hip/hip_runtime.h>
#include <math.h>

typedef __attribute__((ext_vector_type(2))) float v2f;
typedef __attribute__((ext_vector_type(8))) float v8f;
typedef __attribute__((ext_vector_type(4))) int   v4i;

#define GXD 256
#define GYD 256
#define GZD 32
#define NC  21
#define COLSZ (GZD * NC)            // 672 floats per staged column
#define NCOLS 18                    // 3 x-neighbors * 6 y-neighbors

#if __has_builtin(__builtin_amdgcn_global_load_async_to_lds_b128)
#define HAVE_ASYNC_LDS 1
#else
#define HAVE_ASYNC_LDS 0
#endif

// ------------------------------------------------------------------
// Kernel 1: zero the update accumulation buffer (176 MB in d_ws)
// ------------------------------------------------------------------
__global__ void zero_update4(float4* __restrict__ upd, int n4) {
  int i = blockIdx.x * blockDim.x + threadIdx.x;
  int stride = gridDim.x * blockDim.x;
  float4 z = make_float4(0.f, 0.f, 0.f, 0.f);
  for (; i < n4; i += stride) upd[i] = z;
}

// ------------------------------------------------------------------
// Kernel 2: voxelize + scatter-add points (global_atomic_add_f32)
// ------------------------------------------------------------------
__global__ void scatter_points(const float* __restrict__ pc,
                               float* __restrict__ upd, int n) {
  int i = blockIdx.x * blockDim.x + threadIdx.x;
  if (i >= n) return;
  float px = pc[i * 4 + 0];
  float py = pc[i * 4 + 1];
  float pz = pc[i * 4 + 2];
  int lab = (int)pc[i * 4 + 3];
  bool valid = (px >= -25.6f) && (px < 25.6f) &&
               (py >= -25.6f) && (py < 25.6f) &&
               (pz >= -2.0f)  && (pz < 1.2f);
  if (!valid) return;  // reference adds 0.0 for invalid -> no-op
  const float vx = 51.2f / 256.0f, vy = 51.2f / 256.0f, vz = 3.2f / 32.0f;
  int ix = (int)floorf((px + 25.6f) / vx);
  int iy = (int)floorf((py + 25.6f) / vy);
  int iz = (int)floorf((pz + 2.0f) / vz);
  ix = min(max(ix, 0), GXD - 1);
  iy = min(max(iy, 0), GYD - 1);
  iz = min(max(iz, 0), GZD - 1);
  lab = min(max(lab, 0), NC - 1);
  int idx = (((ix * GYD) + iy) * GZD + iz) * NC + lab;
  atomicAdd(&upd[idx], 1.0f);
}

// ------------------------------------------------------------------
// Kernel 3: banded-GEMM 3x3x3 conv via V_WMMA_F32_16X16X4_F32 + add.
// Block = 128 threads = 4 waves; each wave owns one (x,y) column
// (32 z x 21 classes). The z-conv is a 32x32 tridiagonal matrix per
// (dx,dy) neighbor: out_col += T_j (32x32 banded) * U_j (32x21).
// D tiles: 2 z-blocks x 2 class-blocks; K walked in chunks of 4,
// only 5 chunks per z-block intersect the +/-1 band. Band indicator
// masks are hoisted out of the 9-neighbor loop (A via packed FMAs).
// ------------------------------------------------------------------
__global__ __launch_bounds__(128) void conv_add(
    const float* __restrict__ cm, const float* __restrict__ upd,
    const float* __restrict__ w, float* __restrict__ out) {
  __shared__ float lds[NCOLS * COLSZ + 16];   // +tail pad for stray B reads

  const int tid = threadIdx.x;
  const int x   = blockIdx.x;
  const int y0  = blockIdx.y * 4;

  // ---- stage halo columns (aligned float4 / async-to-LDS), zero-pad edges
  for (int t = tid; t < NCOLS * (COLSZ / 4); t += 128) {
    int col = t / (COLSZ / 4);
    int q   = t - col * (COLSZ / 4);
    int gx  = x  + (col / 6) - 1;
    int gy  = y0 + (col % 6) - 1;
    float4* dst = (float4*)&lds[col * COLSZ + q * 4];
    if (gx >= 0 && gx < GXD && gy >= 0 && gy < GYD) {
      const float4* src = (const float4*)(upd + ((gx * GYD) + gy) * GZD * NC) + q;
#if HAVE_ASYNC_LDS
      __builtin_amdgcn_global_load_async_to_lds_b128((v4i*)src, (v4i*)dst, 0, 0);
#else
      *dst = *src;
#endif
    } else {
      *dst = make_float4(0.f, 0.f, 0.f, 0.f);
    }
  }
#if HAVE_ASYNC_LDS
#if __has_builtin(__builtin_amdgcn_s_wait_asynccnt)
  __builtin_amdgcn_s_wait_asynccnt(0);
#else
  asm volatile("s_wait_asynccnt 0x0" ::: "memory");
#endif
#endif
  __syncthreads();

  // ---- per-thread filter: sigmoid(w), center forced to 1.0.
  //      exp scalarizes (w uniform); use v_rcp_f32 (1 ulp) instead of
  //      the full IEEE divide expansion.
  float f[27];
#pragma unroll
  for (int k = 0; k < 27; ++k)
    f[k] = __builtin_amdgcn_rcpf(1.0f + expf(-w[k]));
  f[13] = 1.0f;  // (kx,ky,kz) = (1,1,1)

  const int wv   = tid >> 5;   // wave id -> y column
  const int lane = tid & 31;
  const int h    = lane >> 4;  // lane half
  const int l15  = lane & 15;
  const int yy   = y0 + wv;

#pragma unroll
  for (int m = 0; m < 2; ++m) {   // z-blocks: rows z0..z0+15
    const int z0 = m * 16;
    const int zA = z0 + l15;      // A-matrix row (M) for this lane
    v8f acc0 = {};                // classes 0..15
    v8f acc1 = {};                // classes 16..31 (>=21 discarded)
#pragma unroll
    for (int cki = 0; cki < 5; ++cki) {
      const int ck  = (m == 0) ? cki : (3 + cki);  // K chunk of 4
      const int zp0 = ck * 4 + 2 * h;              // A/B K index: 2h+v
      const int d0  = zp0 - zA;
      const int d1  = d0 + 1;
      // band indicator masks (shared by all 9 neighbors)
      const float p0x = (d0 == -1) ? 1.f : 0.f;
      const float p1x = (d0 ==  0) ? 1.f : 0.f;
      const float p2x = (d0 ==  1) ? 1.f : 0.f;
      const float p0y = (d1 == -1) ? 1.f : 0.f;
      const float p1y = (d1 ==  0) ? 1.f : 0.f;
      const float p2y = (d1 ==  1) ? 1.f : 0.f;
      const int zoff = zp0 * NC;
#pragma unroll
      for (int jx = 0; jx < 3; ++jx) {
#pragma unroll
        for (int jy = 0; jy < 3; ++jy) {
          const float fj0 = f[(jx * 3 + jy) * 3 + 0];  // tap z-1
          const float fj1 = f[(jx * 3 + jy) * 3 + 1];  // tap z
          const float fj2 = f[(jx * 3 + jy) * 3 + 2];  // tap z+1
          v2f a;
          a.x = fj0 * p0x + fj1 * p1x + fj2 * p2x;
          a.y = fj0 * p0y + fj1 * p1y + fj2 * p2y;
          const int colbase = (jx * 6 + (wv + jy)) * COLSZ;
          v2f b0, b1;
          b0.x = lds[colbase + zoff + l15];
          b0.y = lds[colbase + zoff + NC + l15];
          b1.x = lds[colbase + zoff + 16 + l15];
          b1.y = lds[colbase + zoff + NC + 16 + l15];
          acc0 = __builtin_amdgcn_wmma_f32_16x16x4_f32(
              false, a, false, b0, (short)0, acc0, false, false);
          acc1 = __builtin_amdgcn_wmma_f32_16x16x4_f32(
              false, a, false, b1, (short)0, acc1, false, false);
        }
      }
    }
    // ---- epilogue: D layout M = z0 + 8h + r, N = l15 (+16 for acc1)
#pragma unroll
    for (int r = 0; r < 8; ++r) {
      const int z = z0 + 8 * h + r;
      const int base = (((x * GYD) + yy) * GZD + z) * NC;
      out[base + l15] = cm[base + l15] + acc0[r];
      if (l15 < NC - 16)
        out[base + 16 + l15] = cm[base + 16 + l15] + acc1[r];
    }
  }
}

// ------------------------------------------------------------------
extern "C" void kernel_launch(void* const* d_in, const int* in_sizes, int n_in,
                              void* d_out, int out_size, void* d_ws, size_t ws_size,
                              hipStream_t stream) {
  const float* cm = (const float*)d_in[0];  // current_map (256,256,32,21)
  const float* pc = (const float*)d_in[1];  // point_cloud (1e6,4)
  const float* w  = (const float*)d_in[2];  // weights (27,)
  float* out = (float*)d_out;
  float* upd = (float*)d_ws;                // 44,040,192 f32 scratch

  const int n  = GXD * GYD * GZD * NC;      // 44,040,192
  const int n4 = n / 4;
  zero_update4<<<dim3(2048), dim3(256), 0, stream>>>((float4*)upd, n4);

  const int npts = in_sizes[1] / 4;
  scatter_points<<<dim3((npts + 255) / 256), dim3(256), 0, stream>>>(pc, upd, npts);

  conv_add<<<dim3(GXD, GYD / 4), dim3(128), 0, stream>>>(cm, upd, w, out);
}